// LovaszLoss_63848983823243
// MI455X (gfx1250) — compile-verified
//
#include <hip/hip_runtime.h>
#include <hip/hip_bf16.h>
#include <stdint.h>

// ---------------------------------------------------------------------------
// Lovász hinge loss, B=16 C=4 H=256 W=1600  ->  64 channels x 409600 elems.
//   1) transform: error -> descending-order uint32 key; split by label into
//      pos/neg regions (key-only sort: equal-error ties are order-invariant).
//   2) 4-pass LSD radix sort per segment; stable wave32 ballot match-any
//      ranking; two-phase LDS scatter for coalesced global writes; per-tile
//      digit bases fetched by the Tensor Data Mover (tensor_load_to_lds).
//   3) merge-path merge of posSorted/negSorted with async LDS staging
//      (global_load_async_to_lds_b32 + s_wait_asynccnt); exact integer
//      jaccard counts; deterministic tree reductions (no float atomics).
//   4) final mean over 64 channels via V_WMMA_F32_16X16X4_F32 (ones matrix).
// ---------------------------------------------------------------------------

#define PCH 409600          // elements per channel (H*W)
#define NCH 64              // B*C channels
#define TPC 100             // tiles per channel (PCH / 4096)
#define TILE 4096
#define VT 16

typedef __attribute__((ext_vector_type(2))) float v2f;
typedef __attribute__((ext_vector_type(8))) float v8f;
typedef __attribute__((ext_vector_type(4))) unsigned v4u;
typedef __attribute__((ext_vector_type(8))) unsigned v8u;

// ---------------- wave32 helpers ----------------
__device__ __forceinline__ float wave_reduce_f32(float v, unsigned lane) {
#pragma unroll
  for (int o = 16; o > 0; o >>= 1) {
    int iv = __builtin_amdgcn_ds_bpermute((int)(((lane ^ (unsigned)o) & 31u) << 2),
                                          __float_as_int(v));
    v += __int_as_float(iv);
  }
  return v;
}
__device__ __forceinline__ unsigned wave_incl_scan_u32(unsigned v, unsigned lane) {
#pragma unroll
  for (int o = 1; o < 32; o <<= 1) {
    unsigned y = (unsigned)__builtin_amdgcn_ds_bpermute(
        (int)(((lane - (unsigned)o) & 31u) << 2), (int)v);
    if (lane >= (unsigned)o) v += y;
  }
  return v;
}

// float error -> key such that ascending uint order == descending error order
__device__ __forceinline__ uint32_t err_to_key(float e) {
  uint32_t u = __float_as_uint(e);
  uint32_t kasc = u ^ (uint32_t)(((int32_t)u >> 31) | 0x80000000);
  return ~kasc;
}
__device__ __forceinline__ float key_to_err(uint32_t key) {
  uint32_t kasc = ~key;
  uint32_t u = (kasc & 0x80000000u) ? (kasc ^ 0x80000000u) : ~kasc;
  return __uint_as_float(u);
}
// low 32 bits of a flat pointer into LDS == wave-relative LDS byte address
__device__ __forceinline__ uint32_t lds_addr32(const void* p) {
  return (uint32_t)(uintptr_t)p;
}

// ---------------- kernels ----------------

__global__ void k_init(uint32_t* cnt) {
  if (threadIdx.x < 128) cnt[threadIdx.x] = 0u;
}

__global__ __launch_bounds__(256) void k_transform(
    const float* __restrict__ logits, const int* __restrict__ labels,
    uint32_t* __restrict__ posOut, uint32_t* __restrict__ negOut,
    uint32_t* __restrict__ cntPos, uint32_t* __restrict__ cntNeg, int chBase) {
  int wg = blockIdx.x;
  int chLocal = wg / TPC;
  int chGlobal = chBase + chLocal;
  int p0 = (wg % TPC) * TILE;
  int tid = threadIdx.x;
  unsigned lane = tid & 31u;
  int wave = tid >> 5;

  const float4* src4 = (const float4*)(logits + (size_t)chGlobal * PCH + p0);
  const int4* lab4 = (const int4*)(labels + (size_t)chGlobal * PCH + p0);

  uint32_t keys[VT];
  unsigned pmask = 0u;
#pragma unroll
  for (int r = 0; r < 4; ++r) {
    float4 f = src4[r * 256 + tid];     // global_load_b128
    int4 tv = lab4[r * 256 + tid];
    float fe[4] = {f.x, f.y, f.z, f.w};
    int tt[4] = {tv.x, tv.y, tv.z, tv.w};
#pragma unroll
    for (int j = 0; j < 4; ++j) {
      float e = tt[j] ? (1.0f - fe[j]) : (1.0f + fe[j]);
      keys[r * 4 + j] = err_to_key(e);
      if (tt[j]) pmask |= (1u << (r * 4 + j));
    }
  }
  unsigned cpos = (unsigned)__builtin_popcount(pmask);
  unsigned incl = wave_incl_scan_u32(cpos, lane);

  __shared__ unsigned wposTot[8];
  __shared__ unsigned basePosS, baseNegS;
  if (lane == 31u) wposTot[wave] = incl;
  __syncthreads();
  unsigned waveBase = 0u, totPos = 0u;
#pragma unroll
  for (int w = 0; w < 8; ++w) {
    unsigned v = wposTot[w];
    if (w < wave) waveBase += v;
    totPos += v;
  }
  if (tid == 0) {
    basePosS = atomicAdd(&cntPos[chGlobal], totPos);
    baseNegS = atomicAdd(&cntNeg[chGlobal], (unsigned)TILE - totPos);
  }
  __syncthreads();

  unsigned posStart = basePosS + waveBase + incl - cpos;
  unsigned negStart = baseNegS + (unsigned)tid * VT - (waveBase + incl - cpos);
  uint32_t* pdst = posOut + (size_t)chLocal * PCH;
  uint32_t* ndst = negOut + (size_t)chLocal * PCH;
  unsigned pc = 0u, nc = 0u;
#pragma unroll
  for (int i = 0; i < VT; ++i) {
    if (pmask & (1u << i)) pdst[posStart + pc++] = keys[i];
    else                   ndst[negStart + nc++] = keys[i];
  }
}

__global__ __launch_bounds__(256) void k_hist(
    const uint32_t* __restrict__ keys, const uint32_t* __restrict__ cnt,
    uint32_t* __restrict__ tileHist, int pass, int chBase) {
  __shared__ unsigned h[256];
  int wg = blockIdx.x, tid = threadIdx.x;
  int sLocal = wg / TPC, t = wg % TPC;
  unsigned n = cnt[chBase + sLocal];
  unsigned t0 = (unsigned)t * TILE;
  unsigned lim = (n > t0) ? ((n - t0 < TILE) ? (n - t0) : (unsigned)TILE) : 0u;
  const uint32_t* src = keys + (size_t)sLocal * PCH + t0;
  h[tid] = 0u;
  __syncthreads();
  int sh = pass * 8;
  if (lim == (unsigned)TILE) {
    const uint4* s4 = (const uint4*)src;
#pragma unroll
    for (int r = 0; r < 4; ++r) {
      uint4 v = s4[r * 256 + tid];     // global_load_b128
      atomicAdd(&h[(v.x >> sh) & 255u], 1u);
      atomicAdd(&h[(v.y >> sh) & 255u], 1u);
      atomicAdd(&h[(v.z >> sh) & 255u], 1u);
      atomicAdd(&h[(v.w >> sh) & 255u], 1u);
    }
  } else {
#pragma unroll
    for (int i = 0; i < VT; ++i) {
      unsigned idx = (unsigned)(i * 256 + tid);
      if (idx < lim) atomicAdd(&h[(src[idx] >> sh) & 255u], 1u);
    }
  }
  __syncthreads();
  tileHist[(size_t)wg * 256 + tid] = h[tid];
}

__global__ __launch_bounds__(256) void k_scan(uint32_t* __restrict__ tileHist) {
  int s = blockIdx.x, d = threadIdx.x;
  __shared__ unsigned tot[256];
  size_t base = (size_t)s * TPC * 256;
  unsigned acc = 0u;
  for (int t = 0; t < TPC; ++t) acc += tileHist[base + (size_t)t * 256 + d];
  tot[d] = acc;
  __syncthreads();
  unsigned own = acc;
  for (int o = 1; o < 256; o <<= 1) {
    unsigned y = (d >= o) ? tot[d - o] : 0u;
    __syncthreads();
    tot[d] += y;
    __syncthreads();
  }
  unsigned run = tot[d] - own;  // exclusive digit base within segment
  for (int t = 0; t < TPC; ++t) {
    unsigned v = tileHist[base + (size_t)t * 256 + d];
    tileHist[base + (size_t)t * 256 + d] = run;
    run += v;
  }
}

__global__ __launch_bounds__(256) void k_scatter(
    const uint32_t* __restrict__ in, uint32_t* __restrict__ out,
    const uint32_t* __restrict__ cnt, const uint32_t* __restrict__ tileHist,
    int pass, int chBase) {
  __shared__ unsigned run[256];       // running local digit offsets / counts
  __shared__ int combo[256];          // globalBase[d] - localExclStart[d]
  __shared__ unsigned gbase[256];     // absolute digit bases (TDM-loaded)
  __shared__ unsigned wdc[8 * 256];
  __shared__ uint32_t keyBuf[TILE];
  int wg = blockIdx.x, tid = threadIdx.x;
  unsigned lane = tid & 31u;
  int wave = tid >> 5;
  int sLocal = wg / TPC, t = wg % TPC;
  unsigned n = cnt[chBase + sLocal];
  unsigned t0 = (unsigned)t * TILE;
  if (t0 >= n) return;  // uniform early-out (before any barrier)
  unsigned lim = (n - t0 < TILE) ? (n - t0) : (unsigned)TILE;

  // ---- Tensor Data Mover: fetch this tile's 256 absolute digit bases into
  // LDS. 1-D D#: 256 x 4B, issued once (wave 0), waited just before use.
  if (wave == 0) {
    uint64_t ga = (uint64_t)(uintptr_t)(tileHist + (size_t)wg * 256);
    v4u g0;
    g0[0] = 1u;                                   // count=1, user descriptor
    g0[1] = lds_addr32(&gbase[0]);                // lds_addr
    g0[2] = (uint32_t)ga;                         // global_addr[31:0]
    g0[3] = (uint32_t)((ga >> 32) & 0x01FFFFFFull) | (2u << 30);  // type=2
    v8u g1;
    g1[0] = (2u << 16);            // data_size=4B; no multicast/pad/iterate
    g1[1] = (256u << 16);          // tensor_dim0[15:0] @bit48
    g1[2] = (1u << 16);            // tensor_dim0[31:16]=0 | tensor_dim1=1
    g1[3] = (256u << 16);          // tensor_dim1 hi=0 | tile_dim0=256
    g1[4] = 0u;                    // tile_dim1=0, tile_dim2=0 (1-D tile)
    g1[5] = 256u;                  // tensor_dim0_stride
    g1[6] = 0u;
    g1[7] = 0u;
    asm volatile("tensor_load_to_lds %0, %1" :: "s"(g0), "s"(g1) : "memory");
  }

  const uint32_t* src = in + (size_t)sLocal * PCH + t0;
  run[tid] = 0u;
  int sh = pass * 8;
  uint32_t ltmask = (lane == 0u) ? 0u : (0xFFFFFFFFu >> (32u - lane));

  // ---- phase A: stable local ranking into LDS keyBuf ----
  for (int r = 0; r < VT; ++r) {
    unsigned li = (unsigned)(r * 256 + tid);
    bool active = li < lim;
    uint32_t key = active ? src[li] : 0u;
    unsigned digit = (key >> sh) & 255u;
#pragma unroll
    for (int w = 0; w < 8; ++w) wdc[w * 256 + tid] = 0u;
    __syncthreads();
    // wave32 match-any over 8 digit bits
    uint32_t peers = __builtin_amdgcn_ballot_w32(active);
#pragma unroll
    for (int b = 0; b < 8; ++b) {
      uint32_t bb = __builtin_amdgcn_ballot_w32(((digit >> b) & 1u) != 0u);
      peers &= ((digit >> b) & 1u) ? bb : ~bb;
    }
    unsigned rankInWave = (unsigned)__builtin_popcount(peers & ltmask);
    bool leader = active && ((peers & ltmask) == 0u);
    if (leader) wdc[wave * 256 + digit] = (unsigned)__builtin_popcount(peers);
    __syncthreads();
    unsigned s = 0u;  // thread tid scans digit column tid across 8 waves
#pragma unroll
    for (int w = 0; w < 8; ++w) {
      unsigned v = wdc[w * 256 + tid];
      wdc[w * 256 + tid] = s;
      s += v;
    }
    __syncthreads();
    if (active)
      keyBuf[run[digit] + wdc[wave * 256 + digit] + rankInWave] = key;
    __syncthreads();
    run[tid] += s;
    __syncthreads();
  }

  // run[d] == tile count of digit d; exclusive scan -> local digit starts
  unsigned tot = run[tid];
  for (int o = 1; o < 256; o <<= 1) {
    unsigned y = (tid >= o) ? run[tid - o] : 0u;
    __syncthreads();
    run[tid] += y;
    __syncthreads();
  }
  unsigned excl = run[tid] - tot;

  if (wave == 0) __builtin_amdgcn_s_wait_tensorcnt(0);  // TDM data landed
  __syncthreads();
  combo[tid] = (int)gbase[tid] - (int)excl;
  __syncthreads();

  // ---- phase B: coalesced digit-run writes ----
  uint32_t* dst = out + (size_t)sLocal * PCH;
  for (unsigned i = (unsigned)tid; i < lim; i += 256u) {
    uint32_t k = keyBuf[i];
    unsigned d = (k >> sh) & 255u;
    dst[(unsigned)(combo[d] + (int)i)] = k;
  }
}

__global__ void k_partition(const uint32_t* __restrict__ posK,
                            const uint32_t* __restrict__ negK,
                            const uint32_t* __restrict__ cntPos,
                            uint32_t* __restrict__ splits, int chBase, int G) {
  int gid = blockIdx.x * blockDim.x + threadIdx.x;
  int chLocal = gid / (TPC + 1), bidx = gid % (TPC + 1);
  if (chLocal >= G) return;
  unsigned np = cntPos[chBase + chLocal];
  unsigned nn = (unsigned)PCH - np;
  const uint32_t* a = posK + (size_t)chLocal * PCH;
  const uint32_t* b = negK + (size_t)chLocal * PCH;
  unsigned d = (unsigned)bidx * TILE;
  unsigned lo = (d > nn) ? (d - nn) : 0u;
  unsigned hi = (d < np) ? d : np;
  while (lo < hi) {
    unsigned m = (lo + hi) >> 1;
    if (b[d - m - 1u] >= a[m]) lo = m + 1u; else hi = m;
  }
  splits[chLocal * (TPC + 1) + bidx] = lo;
}

__global__ __launch_bounds__(256) void k_merge(
    const uint32_t* __restrict__ posK, const uint32_t* __restrict__ negK,
    const uint32_t* __restrict__ cntPos, const uint32_t* __restrict__ splits,
    float* __restrict__ tileLoss, int chBase) {
  __shared__ uint32_t sA[TILE + 1];
  __shared__ uint32_t sB[TILE + 1];
  __shared__ float wsum[8];
  int wg = blockIdx.x, tid = threadIdx.x;
  int chLocal = wg / TPC, t = wg % TPC;
  unsigned np = cntPos[chBase + chLocal];
  const uint32_t* a = posK + (size_t)chLocal * PCH;
  const uint32_t* b = negK + (size_t)chLocal * PCH;
  unsigned i0 = splits[chLocal * (TPC + 1) + t];
  unsigned i1 = splits[chLocal * (TPC + 1) + t + 1];
  unsigned d0 = (unsigned)t * TILE;
  unsigned j0 = d0 - i0, j1 = (d0 + TILE) - i1;
  unsigned aLen = i1 - i0, bLen = j1 - j0;

  // ---- async staging: global -> LDS without VGPR round-trip ----
  uint32_t sAoff = lds_addr32(&sA[0]);
  uint32_t sBoff = lds_addr32(&sB[0]);
  for (unsigned k = (unsigned)tid; k < aLen; k += 256u)
    asm volatile("global_load_async_to_lds_b32 %0, %1, %2"
                 :: "v"(sAoff + (k << 2)), "v"((i0 + k) << 2), "s"(a)
                 : "memory");
  for (unsigned k = (unsigned)tid; k < bLen; k += 256u)
    asm volatile("global_load_async_to_lds_b32 %0, %1, %2"
                 :: "v"(sBoff + (k << 2)), "v"((j0 + k) << 2), "s"(b)
                 : "memory");
  if (tid == 0) { sA[aLen] = 0xFFFFFFFFu; sB[bLen] = 0xFFFFFFFFu; }
  asm volatile("s_wait_asynccnt 0x0" ::: "memory");
  __syncthreads();

  // per-thread diagonal split inside the tile
  unsigned dt = (unsigned)tid * VT;
  unsigned lo = (dt > bLen) ? (dt - bLen) : 0u;
  unsigned hi = (dt < aLen) ? dt : aLen;
  while (lo < hi) {
    unsigned m = (lo + hi) >> 1;
    if (sB[dt - m - 1u] >= sA[m]) lo = m + 1u; else hi = m;
  }
  unsigned i = lo, j = dt - lo;
  float gts = (float)np;
  unsigned kg = d0 + dt, pg = i0 + i;
  float jacPrev = (kg == 0u)
      ? 0.0f
      : (1.0f - (gts - (float)pg) / (gts + (float)(kg - pg)));
  float acc = 0.0f;
#pragma unroll
  for (int s = 0; s < VT; ++s) {
    uint32_t key;
    if (sA[i] <= sB[j]) { key = sA[i]; ++i; ++pg; }
    else                { key = sB[j]; ++j; }
    ++kg;
    float jac = 1.0f - (gts - (float)pg) / (gts + (float)(kg - pg));
    float e = key_to_err(key);
    if (e > 0.0f) acc += e * (jac - jacPrev);
    jacPrev = jac;
  }
  unsigned lane = tid & 31u;
  int wave = tid >> 5;
  acc = wave_reduce_f32(acc, lane);
  if (lane == 0u) wsum[wave] = acc;
  __syncthreads();
  if (tid == 0) {
    float s2 = 0.0f;
#pragma unroll
    for (int w = 0; w < 8; ++w) s2 += wsum[w];
    tileLoss[wg] = s2;
  }
}

__global__ __launch_bounds__(256) void k_group_reduce(
    const float* __restrict__ tileLoss, float* __restrict__ chLoss, int chBase) {
  __shared__ float wsum[8];
  int chLocal = blockIdx.x, tid = threadIdx.x;
  float v = (tid < TPC) ? tileLoss[chLocal * TPC + tid] : 0.0f;
  unsigned lane = tid & 31u;
  int wave = tid >> 5;
  v = wave_reduce_f32(v, lane);
  if (lane == 0u) wsum[wave] = v;
  __syncthreads();
  if (tid == 0) {
    float s = 0.0f;
#pragma unroll
    for (int w = 0; w < 8; ++w) s += wsum[w];
    chLoss[chBase + chLocal] = s;
  }
}

// Final mean over 64 channel losses via V_WMMA_F32_16X16X4_F32:
// A(16x4)=losses, B(4x16)=ones -> D[m][n]=sum_k A[m][k]; total = sum_m D[m][0].
__global__ void k_final(const float* __restrict__ chLoss, float* __restrict__ out) {
  unsigned lane = threadIdx.x & 31u;
  unsigned m = lane & 15u;
  unsigned k0 = (lane < 16u) ? 0u : 2u;
  v2f a, b;
  a.x = chLoss[m * 4u + k0];
  a.y = chLoss[m * 4u + k0 + 1u];
  b.x = 1.0f; b.y = 1.0f;
  v8f c = {0.f, 0.f, 0.f, 0.f, 0.f, 0.f, 0.f, 0.f};
#if __has_builtin(__builtin_amdgcn_wmma_f32_16x16x4_f32)
  c = __builtin_amdgcn_wmma_f32_16x16x4_f32(false, a, false, b, (short)0, c,
                                            false, false);
  float s = c[0] + c[1] + c[2] + c[3] + c[4] + c[5] + c[6] + c[7];
  float s0 = __int_as_float(__builtin_amdgcn_ds_bpermute(0 << 2, __float_as_int(s)));
  float s16 = __int_as_float(__builtin_amdgcn_ds_bpermute(16 << 2, __float_as_int(s)));
  if (lane == 0u) out[0] = (s0 + s16) * (1.0f / 64.0f);
#else
  if (lane == 0u) {
    float tsum = 0.0f;
    for (int i = 0; i < NCH; ++i) tsum += chLoss[i];
    out[0] = tsum * (1.0f / 64.0f);
  }
#endif
}

// ---------------- host side ----------------
extern "C" void kernel_launch(void* const* d_in, const int* in_sizes, int n_in,
                              void* d_out, int out_size, void* d_ws, size_t ws_size,
                              hipStream_t stream) {
  (void)in_sizes; (void)n_in; (void)out_size;
  const float* logits = (const float*)d_in[0];
  const int* labels = (const int*)d_in[1];
  float* out = (float*)d_out;

  auto need = [](int G) -> size_t {
    return 1024ull + 4ull * (size_t)G * PCH * 4ull        // posA negA posB negB
           + (size_t)G * TPC * 256 * 4                    // tileHist
           + (size_t)G * (TPC + 1) * 4                    // splits
           + (size_t)G * TPC * 4;                         // tileLoss
  };
  int G = 64;
  while (G > 1 && need(G) > ws_size) G >>= 1;
  int numGroups = NCH / G;

  char* ws = (char*)d_ws;
  uint32_t* cntPos = (uint32_t*)ws;          // 64
  uint32_t* cntNeg = cntPos + 64;            // 64
  float* chLoss = (float*)(ws + 512);        // 64
  char* p = ws + 1024;
  uint32_t* posA = (uint32_t*)p; p += (size_t)G * PCH * 4;
  uint32_t* negA = (uint32_t*)p; p += (size_t)G * PCH * 4;
  uint32_t* posB = (uint32_t*)p; p += (size_t)G * PCH * 4;
  uint32_t* negB = (uint32_t*)p; p += (size_t)G * PCH * 4;
  uint32_t* tileHist = (uint32_t*)p; p += (size_t)G * TPC * 256 * 4;
  uint32_t* splits = (uint32_t*)p; p += (size_t)G * (TPC + 1) * 4;
  float* tileLoss = (float*)p;

  k_init<<<1, 256, 0, stream>>>(cntPos);  // zeros cntPos+cntNeg (128 u32)

  for (int g = 0; g < numGroups; ++g) {
    int chBase = g * G;
    dim3 tiles(G * TPC);
    k_transform<<<tiles, 256, 0, stream>>>(logits, labels, posA, negA, cntPos,
                                           cntNeg, chBase);
    for (int region = 0; region < 2; ++region) {
      uint32_t* bufA = region ? negA : posA;
      uint32_t* bufB = region ? negB : posB;
      const uint32_t* cnt = region ? cntNeg : cntPos;
      for (int pass = 0; pass < 4; ++pass) {
        uint32_t* src = (pass & 1) ? bufB : bufA;
        uint32_t* dst = (pass & 1) ? bufA : bufB;
        k_hist<<<tiles, 256, 0, stream>>>(src, cnt, tileHist, pass, chBase);
        k_scan<<<G, 256, 0, stream>>>(tileHist);
        k_scatter<<<tiles, 256, 0, stream>>>(src, dst, cnt, tileHist, pass,
                                             chBase);
      }
    }
    int nPart = G * (TPC + 1);
    k_partition<<<(nPart + 127) / 128, 128, 0, stream>>>(posA, negA, cntPos,
                                                         splits, chBase, G);
    k_merge<<<tiles, 256, 0, stream>>>(posA, negA, cntPos, splits, tileLoss,
                                       chBase);
    k_group_reduce<<<G, 256, 0, stream>>>(tileLoss, chLoss, chBase);
  }
  k_final<<<1, 32, 0, stream>>>(chLoss, out);
}